// V3Loss_43499428774004
// MI455X (gfx1250) — compile-verified
//
#include <hip/hip_runtime.h>
#include <hip/hip_bf16.h>
#include <stdint.h>

// ---------------- problem constants ----------------
#define NB      32
#define TOTAL   22743          // 3*(19^2+38^2+76^2)
#define CH      85
#define NTGT    50
#define NCLS    80
#define ROWS    128            // rows per tile/block in the streaming kernel

// per-scale constants
// nH       : 19, 38, 76
// rows/s   : 1083, 4332, 17328
// seg off  : 0, 1083, 5415
// tiles/s  : 9, 34, 136  -> blocks/s: 288, 1088, 4352 -> cum 288, 1376, 5728
#define NBLOCKS 5728
#define REC_F   16             // floats per target record
#define RECS_FLOATS (3*NB*NTGT*REC_F)   // 76800

typedef __attribute__((ext_vector_type(2))) float v2f;
typedef __attribute__((ext_vector_type(8))) float v8f;
typedef int v4i __attribute__((vector_size(16)));   // matches builtin proto

// ---------------- CDNA5 async-to-LDS helpers ----------------
// Builtin protos (from hipcc diagnostics):
//   b128: (v4i __device__* src, v4i __shared__* dst, imm offset, imm cpol)
//   b32 : (int __device__* src, int __shared__* dst, imm offset, imm cpol)
__device__ __forceinline__ void async_b128(const float* g, float* l) {
#if __has_builtin(__builtin_amdgcn_global_load_async_to_lds_b128)
  __builtin_amdgcn_global_load_async_to_lds_b128(
      (__attribute__((address_space(1))) v4i*)g,
      (__attribute__((address_space(3))) v4i*)l, 0, 0);
#else
  l[0] = g[0]; l[1] = g[1]; l[2] = g[2]; l[3] = g[3];
#endif
}

__device__ __forceinline__ void async_b32(const float* g, float* l) {
#if __has_builtin(__builtin_amdgcn_global_load_async_to_lds_b32)
  __builtin_amdgcn_global_load_async_to_lds_b32(
      (__attribute__((address_space(1))) int*)g,
      (__attribute__((address_space(3))) int*)l, 0, 0);
#else
  l[0] = g[0];
#endif
}

__device__ __forceinline__ void wait_async0() {
#if __has_builtin(__builtin_amdgcn_s_wait_asynccnt)
  __builtin_amdgcn_s_wait_asynccnt(0);
#else
  asm volatile("s_wait_asynccnt 0" ::: "memory");
#endif
}

// ---------------- WMMA wave reduction (exact f32) ----------------
// A = [val,0] per lane forms the 16x4 f32 A matrix; B = all-ones 4x16.
// D[m,n] = val[m] + val[m+16]; summing each lane's 8 D regs gives half-sums,
// xor-16 shuffle completes the full 32-lane sum.  EXEC must be all ones.
__device__ __forceinline__ float wave_sum32(float x) {
#if __has_builtin(__builtin_amdgcn_wmma_f32_16x16x4_f32)
  v2f a; a.x = x;   a.y = 0.f;
  v2f b; b.x = 1.f; b.y = 1.f;
  v8f c = {};
  v8f d = __builtin_amdgcn_wmma_f32_16x16x4_f32(false, a, false, b,
                                                (short)0, c, false, false);
  float s = d[0] + d[1] + d[2] + d[3] + d[4] + d[5] + d[6] + d[7];
  s += __shfl_xor(s, 16, 32);
  return s;
#else
  for (int o = 16; o > 0; o >>= 1) x += __shfl_xor(x, o, 32);
  return x;
#endif
}

// ---------------- reference IoU (xywh, centers) ----------------
__device__ __forceinline__ float iou_xywh(float ax, float ay, float aw, float ah,
                                          float bx, float by, float bw, float bh) {
  float x1a = ax - aw * 0.5f, x1b = ax + aw * 0.5f;
  float y1a = ay - ah * 0.5f, y1b = ay + ah * 0.5f;
  float x2a = bx - bw * 0.5f, x2b = bx + bw * 0.5f;
  float y2a = by - bh * 0.5f, y2b = by + bh * 0.5f;
  float iw = fmaxf(fminf(x1b, x2b) - fmaxf(x1a, x2a), 0.f);
  float ih = fmaxf(fminf(y1b, y2b) - fmaxf(y1a, y2a), 0.f);
  float inter = iw * ih;
  float uni = aw * ah + bw * bh - inter;
  return inter / uni;
}

// ---------------- kernel 1: per-target records ----------------
// record (16 floats): [0]=valid [1]=best_idx [2..5]=x,y,w,h (raw fractions)
// [6]=iou_t (tconf) [7]=class idx [8..11]=gt x1,x2,y1,y2 corners [12]=gt area
__global__ void v3loss_prep(const float* __restrict__ output,
                            const float* __restrict__ target,
                            float* __restrict__ recs) {
  const int sb = blockIdx.x;
  const int s = sb / NB, b = sb % NB;
  const int t = threadIdx.x;
  if (t >= NTGT) return;

  const int   nH  = (s == 0) ? 19 : ((s == 1) ? 38 : 76);
  const int   seg = (s == 0) ? 0  : ((s == 1) ? 1083 : 5415);
  const float ANC[3][6] = {
      {3.625f, 2.8125f, 4.875f, 6.1875f, 11.65625f, 10.1875f},
      {1.875f, 3.8125f, 3.875f, 2.8125f, 3.6875f, 7.4375f},
      {1.25f, 1.625f, 2.0f, 3.75f, 4.125f, 2.875f}};

  const float* tg = target + b * (NTGT * 5);
  float fci = tg[t * 5 + 0];
  float x   = tg[t * 5 + 1];
  float y   = tg[t * 5 + 2];
  float w   = tg[t * 5 + 3];
  float h   = tg[t * 5 + 4];

  // valid = cumprod(x != 0) up to and including t
  int valid = 1;
  for (int j = 0; j <= t; ++j) valid &= (tg[j * 5 + 1] != 0.f) ? 1 : 0;

  float gx = x * (float)nH, gy = y * (float)nH;
  float gw = w * (float)nH, gh = h * (float)nH;

  // best anchor by shape IoU (first-max tie-break, like jnp.argmax)
  float bestI = -1.f;
  int   bn    = 0;
  for (int n = 0; n < 3; ++n) {
    float io = iou_xywh(0.f, 0.f, ANC[s][2 * n], ANC[s][2 * n + 1],
                        0.f, 0.f, gw, gh);
    if (io > bestI) { bestI = io; bn = n; }
  }

  int gi = (int)gx, gj = (int)gy;
  int bidx = 3 * (nH * gi + gj) + bn;      // replicates reference's indexing

  const float* p = output + ((size_t)b * TOTAL + seg + bidx) * CH;
  float px = p[0] / 608.f, py = p[1] / 608.f;
  float pw = p[2] / 608.f, ph = p[3] / 608.f;
  float iou_t = iou_xywh(gx, gy, gw, gh, px, py, pw, ph);

  float* rc = recs + ((size_t)((s * NB + b) * NTGT + t)) * REC_F;
  rc[0]  = valid ? 1.f : 0.f;
  rc[1]  = (float)bidx;
  rc[2]  = x;  rc[3] = y;  rc[4] = w;  rc[5] = h;
  rc[6]  = iou_t;
  rc[7]  = fci;
  rc[8]  = gx - gw * 0.5f;  rc[9]  = gx + gw * 0.5f;
  rc[10] = gy - gh * 0.5f;  rc[11] = gy + gh * 0.5f;
  rc[12] = gw * gh;
  rc[13] = 0.f; rc[14] = 0.f; rc[15] = 0.f;
}

// ---------------- kernel 2: streaming loss over all anchors ----------------
__global__ void __launch_bounds__(ROWS)
v3loss_main(const float* __restrict__ output,
            const float* __restrict__ recs,
            float* __restrict__ partials) {
  __shared__ __align__(16) float tile[ROWS * CH];   // 43,520 B
  __shared__ float trec[NTGT * REC_F];              //  3,200 B
  __shared__ float wsum[4];

  const int bid = blockIdx.x, tid = threadIdx.x;

  int s, rb;
  if (bid < 288)        { s = 0; rb = bid; }
  else if (bid < 1376)  { s = 1; rb = bid - 288; }
  else                  { s = 2; rb = bid - 1376; }
  const int tilesPer = (s == 0) ? 9 : ((s == 1) ? 34 : 136);
  const int b  = rb / tilesPer;
  const int tl = rb % tilesPer;
  const int nH  = (s == 0) ? 19 : ((s == 1) ? 38 : 76);
  const int seg = (s == 0) ? 0  : ((s == 1) ? 1083 : 5415);
  const int S   = 3 * nH * nH;
  const int row0  = tl * ROWS;
  const int nrows = (S - row0 < ROWS) ? (S - row0) : ROWS;

  // target records for this (scale,batch) -> LDS (uniform broadcast data)
  const float* rsrc = recs + (size_t)(s * NB + b) * NTGT * REC_F;
  for (int i = tid; i < NTGT * REC_F; i += ROWS) trec[i] = rsrc[i];

  // async DMA of the row tile into LDS (b128 body + b32 ragged tail)
  const float* g = output + ((size_t)b * TOTAL + seg + row0) * CH;
  const int dwords = nrows * CH;
  const int n128 = dwords >> 2, rem = dwords & 3;
  for (int i = tid; i < n128; i += ROWS) async_b128(g + 4 * i, tile + 4 * i);
  if (tid < rem) async_b32(g + 4 * n128 + tid, tile + 4 * n128 + tid);
  wait_async0();
  __syncthreads();

  float sum = 0.f;
  if (tid < nrows) {
    const float* row = &tile[tid * CH];   // stride 85: bank-conflict free
    const int idx = row0 + tid;

    float px = row[0] / 608.f, py = row[1] / 608.f;
    float pw = row[2] / 608.f, ph = row[3] / 608.f;
    float conf = row[5];

    float x1a = px - pw * 0.5f, x1b = px + pw * 0.5f;
    float y1a = py - ph * 0.5f, y1b = py + ph * 0.5f;
    float area1 = pw * ph;

    bool  hi = false, m = false;
    float tx = 0.5f, ty = 0.5f, tw = 0.f, th = 0.f, tc = 0.f;
    unsigned cm0 = 0, cm1 = 0, cm2 = 0;   // 80-bit class scatter mask

    for (int t = 0; t < NTGT; ++t) {
      const float* rc = &trec[t * REC_F];
      if (rc[0] == 0.f) break;            // valid is a cumprod -> monotone
      // iou > 0.6  <=>  inter > 0.6 * union   (division-free predicate)
      float iw = fmaxf(fminf(x1b, rc[9])  - fmaxf(x1a, rc[8]),  0.f);
      float ih = fmaxf(fminf(y1b, rc[11]) - fmaxf(y1a, rc[10]), 0.f);
      float inter = iw * ih;
      float uni   = area1 + rc[12] - inter;
      hi = hi || (inter > 0.6f * uni);
      if ((int)rc[1] == idx) {            // last valid writer wins for values
        m = true;
        tx = rc[2]; ty = rc[3]; tw = rc[4]; th = rc[5]; tc = rc[6];
        int ci = (int)rc[7];              // tcls accumulates ALL class bits
        if (ci < 32) cm0 |= 1u << ci;
        else if (ci < 64) cm1 |= 1u << (ci - 32);
        else cm2 |= 1u << (ci - 64);
      }
    }

    float cmask = m ? 1.f : (hi ? 0.f : 5.f);

    // coord loss: coord_mask == 1 everywhere in the reference
    float dx = px - tx, dy = py - ty, dw = pw - tw, dh = ph - th;
    sum += 0.5f * (dx * dx + dy * dy + dw * dw + dh * dh);

    // conf loss: 0.5 * (m*(p - t))^2
    float dc = cmask * (conf - tc);
    sum += 0.5f * dc * dc;

    // class BCE: bulk term is -log(1-v); correct the scattered classes
    float cl = 0.f;
    for (int c = 0; c < NCLS; ++c) {
      float v = row[5 + c];
      cl -= fmaxf(__logf(1.f - v), -100.f);
    }
    if (m) {
      auto fix = [&](unsigned msk, int base) {
        while (msk) {
          int c = base + __builtin_ctz(msk);
          msk &= msk - 1;
          float v = row[5 + c];
          cl += fmaxf(__logf(1.f - v), -100.f);   // undo bulk term
          cl -= fmaxf(__logf(v), -100.f);         // apply -log p
        }
      };
      fix(cm0, 0); fix(cm1, 32); fix(cm2, 64);
    }
    sum += cl;
  }

  // per-wave exact-f32 reduction via v_wmma_f32_16x16x4_f32 (EXEC all ones)
  float ws = wave_sum32(sum);
  const int lane = tid & 31, wv = tid >> 5;
  if (lane == 0) wsum[wv] = ws;
  __syncthreads();
  if (tid == 0) partials[bid] = wsum[0] + wsum[1] + wsum[2] + wsum[3];
}

// ---------------- kernel 3: deterministic final reduction ----------------
__global__ void v3loss_reduce(const float* __restrict__ partials,
                              float* __restrict__ out) {
  __shared__ float sm[256];
  const int tid = threadIdx.x;
  float a = 0.f;
  for (int i = tid; i < NBLOCKS; i += 256) a += partials[i];
  sm[tid] = a;
  __syncthreads();
  for (int o = 128; o > 0; o >>= 1) {
    if (tid < o) sm[tid] += sm[tid + o];
    __syncthreads();
  }
  if (tid == 0) out[0] = sm[0];
}

// ---------------- launcher ----------------
extern "C" void kernel_launch(void* const* d_in, const int* in_sizes, int n_in,
                              void* d_out, int out_size, void* d_ws, size_t ws_size,
                              hipStream_t stream) {
  const float* output = (const float*)d_in[0];   // (32, 22743, 85) f32
  const float* target = (const float*)d_in[1];   // (32, 250) f32

  float* recs     = (float*)d_ws;                // 76,800 floats
  float* partials = recs + RECS_FLOATS;          //  5,728 floats

  v3loss_prep  <<<3 * NB, 64,  0, stream>>>(output, target, recs);
  v3loss_main  <<<NBLOCKS, ROWS, 0, stream>>>(output, recs, partials);
  v3loss_reduce<<<1, 256, 0, stream>>>(partials, (float*)d_out);
}